// IRNN_71889162600450
// MI455X (gfx1250) — compile-verified
//
#include <hip/hip_runtime.h>

// ---------------------------------------------------------------------------
// IRNN on MI455X (gfx1250), fp32 end-to-end via V_WMMA_F32_16X16X4_F32.
//
// Phase 1: x_proj = inputs @ Wi^T + bi  -> written into d_out[t] slots.
//          One wave computes a 16x64 strip (4 accumulators, A reused 4x).
// Phase 2: persistent scan kernel, 64 WGs x 32 Wh columns; the Wh slice
//          (32 x 2052 f32 ~ 257 KB) is staged into LDS once with
//          GLOBAL_LOAD_ASYNC_TO_LDS_B128 and reused for all 512 steps.
//          h state double-buffers in d_ws; device-wide barrier per step.
// ---------------------------------------------------------------------------

typedef float v2f __attribute__((ext_vector_type(2)));
typedef float v8f __attribute__((ext_vector_type(8)));

#define T_DIM  512
#define B_DIM  64
#define IN_DIM 1024
#define H_DIM  2048

#define NWG2    64                  // persistent workgroups in phase 2
#define NSLICE  32                  // output columns per workgroup (2 N-tiles)
#define LDS_PAD 4                   // row pad (floats) -> conflict-free b64 reads
#define LDS_STRIDE (H_DIM + LDS_PAD)  // 2052 floats per Wh row in LDS

// ---------------------------------------------------------------------------
__global__ void irnn_init_ws(unsigned* counter) {
    if (threadIdx.x == 0 && blockIdx.x == 0) *counter = 0u;
}

// ---------------------------------------------------------------------------
// Phase 1: wave computes 16 rows x 64 cols of x_proj (4 x 16x16 tiles).
// f32 WMMA 16x16x4 fragments: lane L ->
//   A: row m0+(L&15), K = kb + 2*(L>>4) + {0,1}   (contiguous float2)
//   B: col n +(L&15), K = kb + 2*(L>>4) + {0,1}   (contiguous float2, B=Wi^T)
//   C/D: vgpr r, lane L -> (m0 + r + 8*(L>>4), n + (L&15))
__global__ __launch_bounds__(256)
void irnn_xproj(const float* __restrict__ X,   // [T*B, IN]
                const float* __restrict__ Wi,  // [H, IN]
                const float* __restrict__ bi,  // [H]
                float*       __restrict__ out) // [T*B, H] (xp staging)
{
    const int lane = threadIdx.x & 31;
    const int wave = threadIdx.x >> 5;
    const int m0   = blockIdx.x * 16;                  // T*B/16 = 2048
    const int n0   = (blockIdx.y * 8 + wave) * 64;     // grid.y = H/(8*64) = 4
    const int half = lane >> 4;
    const int l16  = lane & 15;

    const float* aptr = X + (size_t)(m0 + l16) * IN_DIM + 2 * half;
    const float* bp0  = Wi + (size_t)(n0 +  0 + l16) * IN_DIM + 2 * half;
    const float* bp1  = Wi + (size_t)(n0 + 16 + l16) * IN_DIM + 2 * half;
    const float* bp2  = Wi + (size_t)(n0 + 32 + l16) * IN_DIM + 2 * half;
    const float* bp3  = Wi + (size_t)(n0 + 48 + l16) * IN_DIM + 2 * half;

    v8f c0 = {}, c1 = {}, c2 = {}, c3 = {};
#pragma unroll 4
    for (int kb = 0; kb < IN_DIM; kb += 4) {
        v2f a  = *(const v2f*)(aptr + kb);
        v2f b0 = *(const v2f*)(bp0 + kb);
        v2f b1 = *(const v2f*)(bp1 + kb);
        v2f b2 = *(const v2f*)(bp2 + kb);
        v2f b3 = *(const v2f*)(bp3 + kb);
        c0 = __builtin_amdgcn_wmma_f32_16x16x4_f32(false, a, false, b0, (short)0, c0, false, false);
        c1 = __builtin_amdgcn_wmma_f32_16x16x4_f32(false, a, false, b1, (short)0, c1, false, false);
        c2 = __builtin_amdgcn_wmma_f32_16x16x4_f32(false, a, false, b2, (short)0, c2, false, false);
        c3 = __builtin_amdgcn_wmma_f32_16x16x4_f32(false, a, false, b3, (short)0, c3, false, false);
    }

    const float bias0 = bi[n0 +  0 + l16];
    const float bias1 = bi[n0 + 16 + l16];
    const float bias2 = bi[n0 + 32 + l16];
    const float bias3 = bi[n0 + 48 + l16];
#pragma unroll
    for (int r = 0; r < 8; ++r) {
        const size_t row = (size_t)(m0 + r + 8 * half) * H_DIM;
        out[row + n0 +  0 + l16] = c0[r] + bias0;
        out[row + n0 + 16 + l16] = c1[r] + bias1;
        out[row + n0 + 32 + l16] = c2[r] + bias2;
        out[row + n0 + 48 + l16] = c3[r] + bias3;
    }
}

// ---------------------------------------------------------------------------
// Phase 2: persistent scan. blockIdx.x owns output columns [n0, n0+32).
// Block = 128 threads = 4 waves; wave w computes batch rows [16w, 16w+16)
// against both 16-column tiles (A fragment reused 2x per k-step).
__global__ __launch_bounds__(128)
void irnn_scan(const float* __restrict__ h0,     // [B, H] initial hidden
               const float* __restrict__ Wh,     // [H, H]
               const float* __restrict__ bh,     // [H]
               float*       __restrict__ out,    // [T*B,H] xp -> h ; tail [B,H] h_last
               float*       __restrict__ hbuf,   // [2, B, H]
               unsigned*    __restrict__ counter)
{
    __shared__ float ldsWh[NSLICE * LDS_STRIDE];   // 32*2052*4 = 262,656 B

    const int tid  = threadIdx.x;
    const int lane = tid & 31;
    const int wave = tid >> 5;
    const int n0   = blockIdx.x * NSLICE;
    const int half = lane >> 4;
    const int l16  = lane & 15;

    // --- stage Wh[n0..n0+31, :] into LDS with async b128 copies ------------
    // Each lane moves 16B per issue; ASYNCcnt tracks completion (no VGPR data).
    {
        const int chunks_per_row = H_DIM / 4;            // 512 x float4
        for (int idx = tid; idx < NSLICE * chunks_per_row; idx += 128) {
            const int nloc = idx / chunks_per_row;
            const int kc   = (idx % chunks_per_row) * 4;
            const float* g = Wh + (size_t)(n0 + nloc) * H_DIM + kc;
            const unsigned ldsoff =
                (unsigned)(uintptr_t)(&ldsWh[nloc * LDS_STRIDE + kc]);
            asm volatile("global_load_async_to_lds_b128 %0, %1, off"
                         :: "v"(ldsoff), "v"(g) : "memory");
        }
        asm volatile("s_wait_asynccnt 0x0" ::: "memory");
    }
    __syncthreads();

    const float  bias0 = bh[n0 +  0 + l16];
    const float  bias1 = bh[n0 + 16 + l16];
    const float* bptr0 = &ldsWh[(l16 +  0) * LDS_STRIDE + 2 * half];
    const float* bptr1 = &ldsWh[(l16 + 16) * LDS_STRIDE + 2 * half];
    const int    mrow  = wave * 16 + l16;          // A row this lane feeds

    for (int t = 0; t < T_DIM; ++t) {
        const float* hprev = (t == 0) ? h0
                           : hbuf + (size_t)((t - 1) & 1) * (B_DIM * H_DIM);
        const float* aptr  = hprev + (size_t)mrow * H_DIM + 2 * half;

        // Pull next step's xp slice toward this WGP while we crunch this step.
        if (t + 1 < T_DIM && tid < B_DIM) {
            const float* nxt = out + (size_t)(t + 1) * (B_DIM * H_DIM)
                                   + (size_t)tid * H_DIM + n0;
            __builtin_prefetch(nxt, 0, 1);         // -> global_prefetch_b8
        }

        v8f c0 = {}, c1 = {};
#pragma unroll 8
        for (int kb = 0; kb < H_DIM; kb += 4) {
            v2f a  = *(const v2f*)(aptr  + kb);
            v2f b0 = *(const v2f*)(bptr0 + kb);
            v2f b1 = *(const v2f*)(bptr1 + kb);
            c0 = __builtin_amdgcn_wmma_f32_16x16x4_f32(false, a, false, b0, (short)0, c0, false, false);
            c1 = __builtin_amdgcn_wmma_f32_16x16x4_f32(false, a, false, b1, (short)0, c1, false, false);
        }

        // epilogue: + bh + xp (read from out[t]), ReLU, write h_t to both
        // the ping-pong buffer and the outputs slot (overwriting xp).
        float* hout = hbuf + (size_t)(t & 1) * (B_DIM * H_DIM);
        float* orow = out + (size_t)t * (B_DIM * H_DIM);
#pragma unroll
        for (int r = 0; r < 8; ++r) {
            const int m  = wave * 16 + r + 8 * half;
            const int na = n0 + l16;
            const int nb = n0 + 16 + l16;
            float v0 = c0[r] + bias0 + orow[m * H_DIM + na];
            float v1 = c1[r] + bias1 + orow[m * H_DIM + nb];
            v0 = v0 > 0.0f ? v0 : 0.0f;
            v1 = v1 > 0.0f ? v1 : 0.0f;
            hout[m * H_DIM + na] = v0;
            hout[m * H_DIM + nb] = v1;
            orow[m * H_DIM + na] = v0;
            orow[m * H_DIM + nb] = v1;
        }

        // --- device-wide barrier (monotonic counter, release/acquire) ------
        __threadfence();
        __syncthreads();
        if (tid == 0) {
            __hip_atomic_fetch_add(counter, 1u, __ATOMIC_RELEASE,
                                   __HIP_MEMORY_SCOPE_AGENT);
            const unsigned target = (unsigned)(t + 1) * (unsigned)NWG2;
            while (__hip_atomic_load(counter, __ATOMIC_ACQUIRE,
                                     __HIP_MEMORY_SCOPE_AGENT) < target) {
                __builtin_amdgcn_s_sleep(2);
            }
        }
        __syncthreads();
    }

    // --- h_last tail: each WG writes its own column slice ------------------
    const float* hfin = hbuf + (size_t)((T_DIM - 1) & 1) * (B_DIM * H_DIM);
#pragma unroll
    for (int r = 0; r < 8; ++r) {
        const int m = wave * 16 + r + 8 * half;
        out[(size_t)T_DIM * B_DIM * H_DIM + m * H_DIM + n0 + l16] =
            hfin[m * H_DIM + n0 + l16];
        out[(size_t)T_DIM * B_DIM * H_DIM + m * H_DIM + n0 + 16 + l16] =
            hfin[m * H_DIM + n0 + 16 + l16];
    }
}

// ---------------------------------------------------------------------------
extern "C" void kernel_launch(void* const* d_in, const int* in_sizes, int n_in,
                              void* d_out, int out_size, void* d_ws, size_t ws_size,
                              hipStream_t stream) {
    const float* X  = (const float*)d_in[0];  // inputs [T,B,IN]
    const float* h0 = (const float*)d_in[1];  // hidden [1,B,H]
    const float* Wi = (const float*)d_in[2];  // [H,IN]
    const float* bi = (const float*)d_in[3];  // [H]
    const float* Wh = (const float*)d_in[4];  // [H,H]
    const float* bh = (const float*)d_in[5];  // [H]
    float* out  = (float*)d_out;              // [T,B,H] ++ [1,B,H]

    float*    hbuf    = (float*)d_ws;                               // 2*B*H f32
    unsigned* counter = (unsigned*)((char*)d_ws +
                          (size_t)2 * B_DIM * H_DIM * sizeof(float));

    irnn_init_ws<<<1, 32, 0, stream>>>(counter);

    dim3 g1(T_DIM * B_DIM / 16, H_DIM / (8 * 64));   // 2048 x 4
    irnn_xproj<<<g1, 256, 0, stream>>>(X, Wi, bi, out);

    irnn_scan<<<NWG2, 128, 0, stream>>>(h0, Wh, bh, out, hbuf, counter);
}